// VectorQuantizer_86474871537912
// MI455X (gfx1250) — compile-verified
//
#include <hip/hip_runtime.h>

typedef float v2f __attribute__((ext_vector_type(2)));
typedef float v8f __attribute__((ext_vector_type(8)));
typedef unsigned int v4u __attribute__((ext_vector_type(4)));
typedef int v4i __attribute__((ext_vector_type(4)));
typedef int v8i __attribute__((ext_vector_type(8)));

#define DDIM   256      // embedding dim
#define KCODES 1024     // codebook size
#define SPW    1024     // spatial positions per batch image (H*W)
#define NTOT   32768    // flattened vectors
#define QELEMS 8388608  // 32*256*32*32
#define SROWS  256      // spatial positions per block (8 waves x 2 strips x 16)
#define XSTR   264      // x LDS row stride (floats): 256 + 8 pad dwords (TDM pad)
#define SESTR  260      // e LDS row stride (floats): 256 + 4 pad dwords (TDM pad)
#define SXF    (DDIM * XSTR)         // 67584 floats: x tile [d][s]
#define SEF    (32 * SESTR)          // 8320 floats: e tile [code][d] (2 k-tiles)
#define SEHOFF (SXF + SEF)           // 75904: 0.5*||e||^2 table (1024)
#define SREDOFF (SEHOFF + KCODES)    // 76928: per-wave loss partials (8)
#define SBKOFF  (SREDOFF + 8)        // 76936: chosen code per s (256 ints)
#define SMEM_BYTES (SBKOFF * 4 + SROWS * 4)   // 308768 B < 320 KB

__device__ __forceinline__ v8f wmma4(v2f a, v2f b, v8f c) {
  return __builtin_amdgcn_wmma_f32_16x16x4_f32(false, a, false, b, (short)0, c,
                                               false, false);
}

// One 2D TDM tile load: global (row-major, stride0 elements/row) -> LDS with
// per-row padding.  data_size=4B, pad_enable=1.  Groups 2/3 zero (2D tensor).
__device__ __forceinline__ void tdm_load_2d(unsigned lds_off, const void* gptr,
                                            unsigned tdim0, unsigned tdim1,
                                            unsigned stride0,
                                            unsigned tile0, unsigned tile1,
                                            unsigned pad_int, unsigned pad_amt) {
  const unsigned long long ga = (unsigned long long)(uintptr_t)gptr;
  v4u g0;
  g0.x = 1u;                                      // count=1 (valid user D#)
  g0.y = lds_off;                                 // lds_addr (bytes)
  g0.z = (unsigned)ga;                            // global_addr[31:0]
  g0.w = (unsigned)((ga >> 32) & 0x01FFFFFFu) | (2u << 30);  // addr[56:32]|type=2
  v8i g1;
  g1[0] = (int)((2u << 16)            // data_size = 4B
                | (1u << 20)          // pad_enable
                | (pad_int << 22)     // pad interval code (7 => 256 dwords)
                | (pad_amt << 25));   // pad amount code (n => n+1 dwords)
  g1[1] = (int)((tdim0 & 0xFFFFu) << 16);
  g1[2] = (int)(((tdim0 >> 16) & 0xFFFFu) | ((tdim1 & 0xFFFFu) << 16));
  g1[3] = (int)(((tdim1 >> 16) & 0xFFFFu) | ((tile0 & 0xFFFFu) << 16));
  g1[4] = (int)(tile1 & 0xFFFFu);
  g1[5] = (int)stride0;               // tensor_dim0_stride[31:0]
  g1[6] = 0;
  g1[7] = 0;
  const v4i z4 = {0, 0, 0, 0};
  const v8i z8 = {0, 0, 0, 0, 0, 0, 0, 0};
  // 6-arg form on this toolchain: (g0, g1, g2, g3, g4, cpol)
  __builtin_amdgcn_tensor_load_to_lds(g0, g1, z4, z4, z8, 0);
}

// ---------------- kernel 1: ehalf[k] = 0.5 * ||e_k||^2 ----------------
__global__ __launch_bounds__(256) void vq_prep(const float* __restrict__ emb,
                                               float* __restrict__ ehalf) {
  const int k = blockIdx.x * 256 + threadIdx.x;
  if (k < KCODES) {
    const float4* row = reinterpret_cast<const float4*>(emb + k * DDIM);
    float s = 0.f;
#pragma unroll 8
    for (int j = 0; j < DDIM / 4; ++j) {
      const float4 v = row[j];
      s += v.x * v.x + v.y * v.y + v.z * v.z + v.w * v.w;
    }
    ehalf[k] = 0.5f * s;
  }
}

// ------- kernel 2: C[code,s] = E x X^T via WMMA, argmax, gather, ST out -------
__global__ __launch_bounds__(256) void vq_main(const float* __restrict__ lat,
                                               const float* __restrict__ emb,
                                               const float* __restrict__ ehalf,
                                               float* __restrict__ out,
                                               float* __restrict__ partials) {
  extern __shared__ float smem[];
  float* sx   = smem;              // [d][s]  stride XSTR
  float* se   = smem + SXF;        // [code][d] stride SESTR (32 codes)
  float* seh  = smem + SEHOFF;
  float* sred = smem + SREDOFF;
  int*   sbk  = reinterpret_cast<int*>(smem + SBKOFF);

  const int t    = threadIdx.x;
  const int lane = t & 31;
  const int wave = t >> 5;
  const int n15  = lane & 15;
  const int g    = lane >> 4;
  const int nbase = blockIdx.x * SROWS;
  const int b     = nbase >> 10;
  const int sbase = nbase & (SPW - 1);

  // One TDM op stages the whole 256x256 f32 x slab (natural [d][s] layout,
  // rows padded 256->264 dwords by the TDM itself).
  if (wave == 0) {
    tdm_load_2d(0u, lat + (size_t)b * (DDIM * SPW) + sbase,
                /*tdim0=*/SPW, /*tdim1=*/DDIM, /*stride0=*/SPW,
                /*tile0=*/SROWS, /*tile1=*/DDIM, /*pad_int=*/7, /*pad_amt=*/7);
  }
  {  // stage 0.5*||e||^2 table (4 KB) while the TDM runs
    const float4 v = *reinterpret_cast<const float4*>(ehalf + t * 4);
    seh[t * 4 + 0] = v.x; seh[t * 4 + 1] = v.y;
    seh[t * 4 + 2] = v.z; seh[t * 4 + 3] = v.w;
  }

  // Fragment pointers.  A = e (lane: code m = n15, depth = 2g+{0,1});
  // B = x^T (lane: col s = n15, depth rows of sx).
  const float* ae0 = se + n15 * SESTR + 2 * g;        // code tile j=0
  const float* ae1 = ae0 + 16 * SESTR;                // code tile j=1
  const float* bx0 = sx + 2 * g * XSTR + wave * 32 + n15;  // s strip r=0
  const float* bx1 = bx0 + 16;                             // s strip r=1

  float best0 = -3.4028235e38f, best1 = -3.4028235e38f;
  int bk0 = 0, bk1 = 0;

  for (int ktp = 0; ktp < KCODES / 32; ++ktp) {
    __syncthreads();                      // previous e tile fully consumed
    if (wave == 0) {
      tdm_load_2d(SXF * 4u, emb + (size_t)ktp * 32 * DDIM,
                  /*tdim0=*/DDIM, /*tdim1=*/KCODES, /*stride0=*/DDIM,
                  /*tile0=*/DDIM, /*tile1=*/32, /*pad_int=*/7, /*pad_amt=*/3);
      __builtin_amdgcn_s_wait_tensorcnt((short)0);  // also covers x on round 0
    }
    __syncthreads();                      // tiles visible to all 8 waves

    const int kbase = ktp * 32;
    // Init accumulators with -0.5||e_k||^2: final acc == score directly.
    const float4 e0 = *reinterpret_cast<const float4*>(seh + kbase + 8 * g);
    const float4 e1 = *reinterpret_cast<const float4*>(seh + kbase + 8 * g + 4);
    const float4 f0 = *reinterpret_cast<const float4*>(seh + kbase + 16 + 8 * g);
    const float4 f1 = *reinterpret_cast<const float4*>(seh + kbase + 16 + 8 * g + 4);
    v8f acc00, acc10;
    acc00[0] = -e0.x; acc00[1] = -e0.y; acc00[2] = -e0.z; acc00[3] = -e0.w;
    acc00[4] = -e1.x; acc00[5] = -e1.y; acc00[6] = -e1.z; acc00[7] = -e1.w;
    acc10[0] = -f0.x; acc10[1] = -f0.y; acc10[2] = -f0.z; acc10[3] = -f0.w;
    acc10[4] = -f1.x; acc10[5] = -f1.y; acc10[6] = -f1.z; acc10[7] = -f1.w;
    v8f acc01 = acc00, acc11 = acc10;

    // 2x2 register blocking: 4 WMMAs per (2 A + 2 B) fragment loads.
#pragma unroll
    for (int dk = 0; dk < DDIM / 4; ++dk) {
      const int o = dk * 4 * XSTR;
      const v2f a0 = *reinterpret_cast<const v2f*>(ae0 + dk * 4);
      const v2f a1 = *reinterpret_cast<const v2f*>(ae1 + dk * 4);
      v2f b0, b1;
      b0.x = bx0[o]; b0.y = bx0[o + XSTR];
      b1.x = bx1[o]; b1.y = bx1[o + XSTR];
      acc00 = wmma4(a0, b0, acc00);
      acc01 = wmma4(a0, b1, acc01);
      acc10 = wmma4(a1, b0, acc10);
      acc11 = wmma4(a1, b1, acc11);
    }

    // Lane-local argmax; ascending code order keeps first-min tie semantics.
#pragma unroll
    for (int v = 0; v < 8; ++v) {
      const int c = kbase + 8 * g + v;
      if (acc00[v] > best0) { best0 = acc00[v]; bk0 = c; }
      if (acc01[v] > best1) { best1 = acc01[v]; bk1 = c; }
    }
#pragma unroll
    for (int v = 0; v < 8; ++v) {
      const int c = kbase + 16 + 8 * g + v;
      if (acc10[v] > best0) { best0 = acc10[v]; bk0 = c; }
      if (acc11[v] > best1) { best1 = acc11[v]; bk1 = c; }
    }
  }

  // Lanes l and l^16 hold the two code halves of the same s: single merge.
  {
    const float o0 = __shfl_xor(best0, 16, 32);
    const int   k0 = __shfl_xor(bk0, 16, 32);
    if (o0 > best0 || (o0 == best0 && k0 < bk0)) { best0 = o0; bk0 = k0; }
    const float o1 = __shfl_xor(best1, 16, 32);
    const int   k1 = __shfl_xor(bk1, 16, 32);
    if (o1 > best1 || (o1 == best1 && k1 < bk1)) { best1 = o1; bk1 = k1; }
  }
  if (g == 0) {
    const int row0 = wave * 32 + n15;
    const int row1 = row0 + 16;
    sbk[row0] = bk0;
    sbk[row1] = bk1;
    out[1 + QELEMS + nbase + row0] = (float)bk0;   // idx output (as f32)
    out[1 + QELEMS + nbase + row1] = (float)bk1;
  }
  __syncthreads();

  // Gather chosen codes, straight-through output (x + (q-x)), loss partial.
  float lsum = 0.f;
  float* qb = out + 1 + (size_t)b * (DDIM * SPW) + sbase;
  for (int i = t * 4; i < SROWS * DDIM; i += 256 * 4) {
    const int d = i >> 8;
    const int s = i & 255;
    const int4  kk = *reinterpret_cast<const int4*>(sbk + s);
    const float4 x = *reinterpret_cast<const float4*>(sx + d * XSTR + s);
    float4 r;
    const float q0 = emb[kk.x * DDIM + d]; const float u0 = q0 - x.x; lsum += u0 * u0; r.x = x.x + u0;
    const float q1 = emb[kk.y * DDIM + d]; const float u1 = q1 - x.y; lsum += u1 * u1; r.y = x.y + u1;
    const float q2 = emb[kk.z * DDIM + d]; const float u2 = q2 - x.z; lsum += u2 * u2; r.z = x.z + u2;
    const float q3 = emb[kk.w * DDIM + d]; const float u3 = q3 - x.w; lsum += u3 * u3; r.w = x.w + u3;
    *reinterpret_cast<float4*>(qb + d * SPW + s) = r;
  }
#pragma unroll
  for (int off = 16; off >= 1; off >>= 1) lsum += __shfl_xor(lsum, off, 32);
  if (lane == 0) sred[wave] = lsum;
  __syncthreads();
  if (t == 0) {
    float tot = 0.f;
#pragma unroll
    for (int w = 0; w < 8; ++w) tot += sred[w];
    partials[blockIdx.x] = tot;
  }
}

// ---------------- kernel 3: deterministic fixed-order loss reduction ----------------
__global__ __launch_bounds__(128) void vq_finish(const float* __restrict__ partials,
                                                 float* __restrict__ out) {
  __shared__ float sred[4];
  const int t = threadIdx.x;
  float s = partials[t];
#pragma unroll
  for (int off = 16; off >= 1; off >>= 1) s += __shfl_xor(s, off, 32);
  if ((t & 31) == 0) sred[t >> 5] = s;
  __syncthreads();
  if (t == 0) {
    // codebook_loss == commitment_loss numerically -> (1 + 0.25) * mean
    out[0] = (sred[0] + sred[1] + sred[2] + sred[3]) * (1.25f / 8388608.0f);
  }
}

extern "C" void kernel_launch(void* const* d_in, const int* in_sizes, int n_in,
                              void* d_out, int out_size, void* d_ws, size_t ws_size,
                              hipStream_t stream) {
  (void)in_sizes; (void)n_in; (void)out_size; (void)ws_size;
  const float* lat = (const float*)d_in[0];   // [32,256,32,32] f32
  const float* emb = (const float*)d_in[1];   // [1024,256]    f32
  float* out      = (float*)d_out;            // [1 | 8388608 | 32768] f32
  float* ehalf    = (float*)d_ws;             // 1024 f32
  float* partials = ehalf + KCODES;           // 128 f32

  (void)hipFuncSetAttribute(reinterpret_cast<const void*>(vq_main),
                            hipFuncAttributeMaxDynamicSharedMemorySize,
                            (int)SMEM_BYTES);

  vq_prep<<<(KCODES + 255) / 256, 256, 0, stream>>>(emb, ehalf);
  vq_main<<<NTOT / SROWS, 256, SMEM_BYTES, stream>>>(lat, emb, ehalf, out, partials);
  vq_finish<<<1, 128, 0, stream>>>(partials, out);
}